// GCN_bare_7687991460116
// MI455X (gfx1250) — compile-verified
//
#include <hip/hip_runtime.h>

#define N_NODES  1000000
#define N_EDGES  16000000
#define N_FEAT   128
#define N_GRAPHS 8192

typedef __attribute__((ext_vector_type(2))) float v2f;
typedef __attribute__((ext_vector_type(8))) float v8f;

// ---------------- zero scratch (deg, agg, pooled) ----------------
__global__ void k_zero(float* __restrict__ deg, float* __restrict__ agg,
                       float* __restrict__ pooled) {
    int i = blockIdx.x * blockDim.x + threadIdx.x;
    if (i < N_NODES) { deg[i] = 0.0f; agg[i] = 0.0f; }
    if (i < N_GRAPHS) pooled[i] = 0.0f;
}

// ---------------- h = x @ W via V_WMMA_F32_16X16X4_F32 ----------------
// One wave per 16-row tile of x. A: 16x4 f32 slice of x; B: 4x16 with only
// column N=0 holding W[k..k+3]; accumulate over K=128 (32 WMMAs).
// A layout (ISA 7.12.2): lane<16 -> row=lane, K={k,k+1}; lane>=16 -> K={k+2,k+3}.
// D layout: lane gives N, VGPR gives M; column N=0 lives in lane0 (M=0..7)
// and lane16 (M=8..15).
__global__ void k_proj_wmma(const float* __restrict__ x,
                            const float* __restrict__ W,
                            float* __restrict__ h) {
    const int lane = threadIdx.x & 31;
    const int wave = (blockIdx.x * blockDim.x + threadIdx.x) >> 5;
    const int nTiles = N_NODES / 16;           // 62500, exact
    if (wave >= nTiles) return;                // wave-uniform: EXEC stays all-1s
    const int row = lane & 15;
    const int hi  = lane >> 4;                 // K sub-offset selector (0/1)
    const float* xrow = x + (size_t)(wave * 16 + row) * N_FEAT + 2 * hi;
    // CDNA5 global_prefetch_b8: pull next tile's lines toward the WGP
    __builtin_prefetch(xrow + 16 * N_FEAT, 0, 3);
    v8f acc = {};
#pragma unroll 8
    for (int k = 0; k < N_FEAT; k += 4) {
        v2f a = *(const v2f*)(xrow + k);
        float w0 = W[k + 2 * hi];
        float w1 = W[k + 1 + 2 * hi];
        v2f b;
        b.x = (row == 0) ? w0 : 0.0f;          // only column N=0 of B is live
        b.y = (row == 0) ? w1 : 0.0f;
        acc = __builtin_amdgcn_wmma_f32_16x16x4_f32(
            /*neg_a=*/false, a, /*neg_b=*/false, b,
            /*c_mod=*/(short)0, acc, /*reuse_a=*/false, /*reuse_b=*/false);
    }
    if (row == 0) {
        // lane0 -> h[tile*16 + 0..7], lane16 -> h[tile*16 + 8..15]
        *(v8f*)(h + wave * 16 + 8 * hi) = acc;
    }
}

// ---------------- degree over edge targets ----------------
__global__ void k_degree(const long long* __restrict__ ei, float* __restrict__ deg) {
    int i = blockIdx.x * blockDim.x + threadIdx.x;
    if (i < N_EDGES) {
        int d = (int)ei[(size_t)N_EDGES + i];  // row 1 = dst
        atomicAdd(&deg[d], 1.0f);
    }
}

// ---------------- dinv = rsqrt(deg + 1)  (self-loop folded in) ----------------
__global__ void k_dinv(float* __restrict__ degdinv) {
    int i = blockIdx.x * blockDim.x + threadIdx.x;
    if (i < N_NODES) degdinv[i] = rsqrtf(degdinv[i] + 1.0f);
}

// ---------------- edge aggregation: agg[d] += dinv[s]*dinv[d]*h[s] ----------------
__global__ void k_edge_agg(const long long* __restrict__ ei,
                           const float* __restrict__ dinv,
                           const float* __restrict__ h,
                           float* __restrict__ agg) {
    int i = blockIdx.x * blockDim.x + threadIdx.x;
    if (i < N_EDGES) {
        int s = (int)ei[i];                    // row 0 = src
        int d = (int)ei[(size_t)N_EDGES + i];  // row 1 = dst
        atomicAdd(&agg[d], dinv[s] * dinv[d] * h[s]);
    }
}

// ---------------- pool per graph, adding self-loop message ----------------
__global__ void k_pool(const float* __restrict__ agg, const float* __restrict__ dinv,
                       const float* __restrict__ h, const long long* __restrict__ batch,
                       float* __restrict__ pooled) {
    int i = blockIdx.x * blockDim.x + threadIdx.x;
    if (i < N_NODES) {
        float v = agg[i] + dinv[i] * dinv[i] * h[i];
        atomicAdd(&pooled[(int)batch[i]], v);
    }
}

// ---------------- out = pooled * pp_w + pp_b ----------------
__global__ void k_final(const float* __restrict__ pooled, const float* __restrict__ pw,
                        const float* __restrict__ pb, float* __restrict__ out) {
    int i = blockIdx.x * blockDim.x + threadIdx.x;
    if (i < N_GRAPHS) out[i] = pooled[i] * pw[0] + pb[0];
}

extern "C" void kernel_launch(void* const* d_in, const int* in_sizes, int n_in,
                              void* d_out, int out_size, void* d_ws, size_t ws_size,
                              hipStream_t stream) {
    const float*     x     = (const float*)d_in[0];      // [1M,128] f32
    const float*     W     = (const float*)d_in[1];      // [128,1]  f32
    const float*     pw    = (const float*)d_in[2];      // [1,1]    f32
    const float*     pb    = (const float*)d_in[3];      // [1]      f32
    const long long* ei    = (const long long*)d_in[4];  // [2,16M]  i64
    const long long* batch = (const long long*)d_in[5];  // [1M]     i64
    float*           out   = (float*)d_out;              // [8192]   f32

    char* ws = (char*)d_ws;
    float* h      = (float*)(ws);               // 4 MB
    float* dinv   = (float*)(ws + 16000000);    // 4 MB (deg, then dinv in place)
    float* agg    = (float*)(ws + 32000000);    // 4 MB
    float* pooled = (float*)(ws + 48000000);    // 32 KB

    const dim3 blk(256);
    k_zero     <<<(N_NODES + 255) / 256,        blk, 0, stream>>>(dinv, agg, pooled);
    k_proj_wmma<<<(N_NODES / 16 + 7) / 8,       blk, 0, stream>>>(x, W, h);
    k_degree   <<<(N_EDGES + 255) / 256,        blk, 0, stream>>>(ei, dinv);
    k_dinv     <<<(N_NODES + 255) / 256,        blk, 0, stream>>>(dinv);
    k_edge_agg <<<(N_EDGES + 255) / 256,        blk, 0, stream>>>(ei, dinv, h, agg);
    k_pool     <<<(N_NODES + 255) / 256,        blk, 0, stream>>>(agg, dinv, h, batch, pooled);
    k_final    <<<(N_GRAPHS + 255) / 256,       blk, 0, stream>>>(pooled, pw, pb, out);
}